// LPVModel2_17669495456432
// MI455X (gfx1250) — compile-verified
//
#include <hip/hip_runtime.h>
#include <math.h>

typedef __attribute__((ext_vector_type(2))) float v2f;
typedef __attribute__((ext_vector_type(8))) float v8f;

#define B_    1024
#define T_    64
#define S_    17
#define C_    16
#define L_    128
#define H_    256
#define N_    128
#define E_    256
#define G_    768          /* 3*H */
#define KPAD  36           /* S+C=33 padded to multiple of 4 */
#define PCOLS (L_*L_ + L_) /* 16512 */
#define TSTEPS 129         /* N+1 trajectory rows */

// ---------------------------------------------------------------------------
// Generic fp32 WMMA GEMM: out[m][n] = sum_k A[m*lda+k] * W(k,n) + bias[n]
//   BT==0: W is (N,K) row-major  -> W[n*ldb + k]   ("nt": x @ W.T)
//   BT==1: W is (K,N) row-major  -> W[k*ldb + n]   ("nn": x @ W)
// One wave per 16x16 tile; 4 waves (128 threads) per block; K multiple of 4.
// A-frag layout (16x4 f32): lanes0-15 K={0,1}, lanes16-31 K={2,3}.
// B-frag layout (4x16 f32): vgpr v holds K = v + 2*(lane>>4), N = lane&15.
// ---------------------------------------------------------------------------
template <int BT>
__global__ __launch_bounds__(128) void gemm16(
    const float* __restrict__ A, int lda,
    const float* __restrict__ W, int ldb,
    const float* __restrict__ bias,
    float* __restrict__ out, int ldc,
    int M, int Ncols, int K)
{
    const int lane = threadIdx.x & 31;
    const int wave = threadIdx.x >> 5;
    const int l16  = lane & 15;
    const int half = lane >> 4;
    const int n0 = (blockIdx.x * 4 + wave) * 16;
    const int m0 = blockIdx.y * 16;
    if (n0 >= Ncols || m0 >= M) return;   // wave-uniform: EXEC stays all-ones for WMMA

    v8f c = {};
    const float* Arow = A + (size_t)(m0 + l16) * lda;
    for (int k0 = 0; k0 < K; k0 += 4) {
        const int ka = k0 + 2 * half;
        v2f a = *(const v2f*)(Arow + ka);
        v2f b;
        if (BT == 0) {
            b = *(const v2f*)(W + (size_t)(n0 + l16) * ldb + ka);
        } else {
            b.x = W[(size_t)ka       * ldb + n0 + l16];
            b.y = W[(size_t)(ka + 1) * ldb + n0 + l16];
        }
        c = __builtin_amdgcn_wmma_f32_16x16x4_f32(false, a, false, b,
                                                  (short)0, c, false, false);
    }
    const float bb = bias[n0 + l16];
#pragma unroll
    for (int r = 0; r < 8; ++r) {
        const int row = m0 + r + 8 * half;   // C layout: vgpr r -> M = r + 8*half
        out[(size_t)row * ldc + n0 + l16] = c[r] + bb;
    }
}

// ---------------------------------------------------------------------------
// Packing / init helpers
// ---------------------------------------------------------------------------
__global__ void pack_ctxt(const float* __restrict__ xh,
                          const float* __restrict__ uh,
                          float* __restrict__ xpad)
{
    size_t idx = (size_t)blockIdx.x * blockDim.x + threadIdx.x;
    if (idx >= (size_t)B_ * T_ * KPAD) return;
    const int j  = (int)(idx % KPAD);
    const size_t bt = idx / KPAD;
    float v = 0.f;
    if (j < S_)            v = xh[bt * S_ + j];
    else if (j < S_ + C_)  v = uh[bt * C_ + (j - S_)];
    xpad[idx] = v;
}

__global__ void pack_wih(const float* __restrict__ wih, float* __restrict__ wihp)
{
    int idx = blockIdx.x * blockDim.x + threadIdx.x;
    if (idx >= G_ * KPAD) return;
    const int j = idx % KPAD, g = idx / KPAD;
    wihp[idx] = (j < S_ + C_) ? wih[g * (S_ + C_) + j] : 0.f;
}

__global__ void zero_f(float* p, int n)
{
    int i = blockIdx.x * blockDim.x + threadIdx.x;
    if (i < n) p[i] = 0.f;
}

// ---------------------------------------------------------------------------
// Fully fused GRU timestep. One 256-thread block owns 16 batches and ALL 768
// gate columns. Both WMMA GEMMs (input part K=36, recurrent part K=256) run
// into separate accumulators; gi/gh pre-activations live only in 96KB of the
// WGP's 320KB LDS (never touch global). After one barrier the same block does
// the sigmoid/tanh gate math and updates its own 16 rows of h in place
// (race-free across blocks: each block reads/writes only rows m0..m0+15).
// 1 launch per timestep instead of 3; saves ~770MB of gi/gh L2 round-trips.
// ---------------------------------------------------------------------------
__global__ __launch_bounds__(256) void gru_step(
    const float* __restrict__ xpad_t,   // xpad + t*KPAD, row stride T_*KPAD
    const float* __restrict__ wihp,     // (768, 36)  nt
    const float* __restrict__ whh,      // (768, 256) nt
    const float* __restrict__ bih,
    const float* __restrict__ bhh,
    float* __restrict__ h)              // (B, 256) updated in place
{
    __shared__ float gi_s[16 * G_];
    __shared__ float gh_s[16 * G_];
    const int tid  = threadIdx.x;
    const int lane = tid & 31, wave = tid >> 5;
    const int l16  = lane & 15, half = lane >> 4;
    const int m0   = blockIdx.x * 16;

    const float* Ax = xpad_t + (size_t)(m0 + l16) * (T_ * KPAD);
    const float* Ah = h      + (size_t)(m0 + l16) * H_;

#pragma unroll 1
    for (int tt = 0; tt < 6; ++tt) {           // 8 waves x 6 tiles = 48 tiles
        const int n0 = (wave + 8 * tt) * 16;
        v8f ci = {}, ch = {};
#pragma unroll
        for (int k0 = 0; k0 < KPAD; k0 += 4) { // input GEMM, K=36
            const int ka = k0 + 2 * half;
            v2f a = *(const v2f*)(Ax + ka);
            v2f b = *(const v2f*)(wihp + (size_t)(n0 + l16) * KPAD + ka);
            ci = __builtin_amdgcn_wmma_f32_16x16x4_f32(false, a, false, b,
                                                       (short)0, ci, false, false);
        }
        for (int k0 = 0; k0 < H_; k0 += 4) {   // recurrent GEMM, K=256
            const int ka = k0 + 2 * half;
            v2f a = *(const v2f*)(Ah + ka);
            v2f b = *(const v2f*)(whh + (size_t)(n0 + l16) * H_ + ka);
            ch = __builtin_amdgcn_wmma_f32_16x16x4_f32(false, a, false, b,
                                                       (short)0, ch, false, false);
        }
        const float bi = bih[n0 + l16], bh = bhh[n0 + l16];
#pragma unroll
        for (int r = 0; r < 8; ++r) {
            const int m = r + 8 * half;
            gi_s[m * G_ + n0 + l16] = ci[r] + bi;
            gh_s[m * G_ + n0 + l16] = ch[r] + bh;
        }
    }
    __syncthreads();   // all reads of h and all LDS writes complete

    // gate phase: thread j handles hidden column j for all 16 batches
    const int j = tid;                         // 0..255 == H_
    for (int m = 0; m < 16; ++m) {
        const float gir = gi_s[m * G_ + j],          ghr = gh_s[m * G_ + j];
        const float giz = gi_s[m * G_ + H_ + j],     ghz = gh_s[m * G_ + H_ + j];
        const float gin = gi_s[m * G_ + 2 * H_ + j], ghn = gh_s[m * G_ + 2 * H_ + j];
        const float r = 1.f / (1.f + __expf(-(gir + ghr)));
        const float z = 1.f / (1.f + __expf(-(giz + ghz)));
        const float n = tanhf(gin + r * ghn);
        const size_t hi = (size_t)(m0 + m) * H_ + j;
        h[hi] = (1.f - z) * n + z * h[hi];
    }
}

// ---------------------------------------------------------------------------
// Build M1 = I - 0.5*dt*A_ct, M2 = I + 0.5*dt*A_ct  (A_ct = W - W^T - diag)
// ---------------------------------------------------------------------------
__global__ void assemble(const float* __restrict__ params,
                         float* __restrict__ M1, float* __restrict__ M2)
{
    size_t idx = (size_t)blockIdx.x * blockDim.x + threadIdx.x;
    if (idx >= (size_t)B_ * L_ * L_) return;
    const int j = (int)(idx % L_);
    const int i = (int)((idx / L_) % L_);
    const int b = (int)(idx / ((size_t)L_ * L_));
    const float* p = params + (size_t)b * PCOLS;
    float a = p[i * L_ + j] - p[j * L_ + i];     // J = W - W^T
    if (i == j && i >= 10) {                     // VERONESE mask
        const float x  = p[L_ * L_ + i];
        const float sp = (x > 20.f) ? x : log1pf(__expf(x));  // softplus
        a -= sp;
    }
    const float halfA = 0.005f * a;              // DT*0.5
    const float eye   = (i == j) ? 1.f : 0.f;
    M1[idx] = eye - halfA;
    M2[idx] = eye + halfA;
}

// ---------------------------------------------------------------------------
// Per-batch Gauss-Jordan: M2 <- M1^{-1} M2 (= A_dt). Both matrices in LDS
// (128KB of the 320KB WGP LDS). Thread j owns column j of both matrices.
// M1 = I - O(dt) is strongly diagonally dominant -> no pivoting needed.
// ---------------------------------------------------------------------------
__global__ __launch_bounds__(128) void solve128(float* __restrict__ M1g,
                                                float* __restrict__ M2g)
{
    __shared__ float m1[L_ * L_];
    __shared__ float m2[L_ * L_];
    __shared__ float fcol[L_];
    const int b = blockIdx.x, j = threadIdx.x;
    const size_t base = (size_t)b * L_ * L_;
    for (int i = 0; i < L_; ++i) {
        m1[i * L_ + j] = M1g[base + i * L_ + j];
        m2[i * L_ + j] = M2g[base + i * L_ + j];
    }
    __syncthreads();
    for (int k = 0; k < L_; ++k) {
        const float pinv = 1.f / m1[k * L_ + k];          // broadcast read
        fcol[j] = (j == k) ? 0.f : m1[j * L_ + k];        // stage pivot column
        const float rk1 = m1[k * L_ + j] * pinv;          // scaled pivot row
        const float rk2 = m2[k * L_ + j] * pinv;
        __syncthreads();                                  // reads done
        m1[k * L_ + j] = rk1;
        m2[k * L_ + j] = rk2;
        for (int i = 0; i < L_; ++i) {                    // rank-1 elimination
            const float f = fcol[i];                      // LDS broadcast
            m1[i * L_ + j] -= f * rk1;
            m2[i * L_ + j] -= f * rk2;
        }
        __syncthreads();
    }
    for (int i = 0; i < L_; ++i)
        M2g[base + i * L_ + j] = m2[i * L_ + j];          // A_dt
}

// ---------------------------------------------------------------------------
// Encoder stage 1 (K=17, too small for WMMA): henc = relu(x_init@enc_w1+b1)
// ---------------------------------------------------------------------------
__global__ void enc1(const float* __restrict__ x_init,
                     const float* __restrict__ w1,
                     const float* __restrict__ b1,
                     float* __restrict__ henc)
{
    int idx = blockIdx.x * blockDim.x + threadIdx.x;
    if (idx >= B_ * E_) return;
    const int b = idx / E_, e = idx % E_;
    float acc = b1[e];
    for (int s = 0; s < S_; ++s) acc += x_init[b * S_ + s] * w1[s * E_ + e];
    henc[idx] = fmaxf(acc, 0.f);
}

// ---------------------------------------------------------------------------
// Persistent rollout: one workgroup per batch. A_dt (64KB, padded to 129
// floats/row for bank-conflict-free column sweeps) + B_mat staged in LDS.
// 128 sequential steps: z <- A_dt z + B_mat u_t.  Bu fused (no 67MB buffer).
// ---------------------------------------------------------------------------
__global__ __launch_bounds__(128) void rollout(const float* __restrict__ Adt,
                                               const float* __restrict__ Bmat,
                                               const float* __restrict__ ufut,
                                               float* __restrict__ ztraj)
{
    __shared__ float As[L_ * (L_ + 1)];   // 128 x 129
    __shared__ float Bs[L_ * 17];         // 128 x 16, padded to 17
    __shared__ float zs[L_];
    const int b = blockIdx.x, i = threadIdx.x;
    const size_t abase = (size_t)b * L_ * L_;
    for (int r = 0; r < L_; ++r)
        As[r * (L_ + 1) + i] = Adt[abase + r * L_ + i];
    for (int cc = 0; cc < C_; ++cc)
        Bs[i * 17 + cc] = Bmat[(size_t)b * L_ * C_ + i * C_ + cc];
    zs[i] = ztraj[(size_t)b * TSTEPS * L_ + i];           // z0
    __syncthreads();
    for (int t = 0; t < N_; ++t) {
        const float* u = ufut + (size_t)b * N_ * C_ + (size_t)t * C_;
        float acc = 0.f;
        for (int jj = 0; jj < L_; ++jj)
            acc += As[i * (L_ + 1) + jj] * zs[jj];        // zs[jj]: broadcast
        for (int cc = 0; cc < C_; ++cc)
            acc += Bs[i * 17 + cc] * u[cc];               // u: L2-cached bcast
        __syncthreads();                                  // old zs fully read
        zs[i] = acc;
        ztraj[(size_t)b * TSTEPS * L_ + (size_t)(t + 1) * L_ + i] = acc;
        __syncthreads();
    }
}

// ---------------------------------------------------------------------------
// Fused decoder MLP: 16 samples/block, 16 waves. Stage1 (128->256) via WMMA
// into LDS (relu), stage2 (256->17) scalar from LDS. No 135MB hidden buffer.
// ---------------------------------------------------------------------------
__global__ __launch_bounds__(512) void decoder(const float* __restrict__ ztraj,
                                               const float* __restrict__ w1,
                                               const float* __restrict__ b1,
                                               const float* __restrict__ w2,
                                               const float* __restrict__ b2,
                                               float* __restrict__ out)
{
    __shared__ float hid[16 * 257];       // 16 samples x 256, padded row 257
    const int tid  = threadIdx.x;
    const int lane = tid & 31, wave = tid >> 5;
    const int l16  = lane & 15, half = lane >> 4;
    const int m0   = blockIdx.x * 16;

    { // stage 1: wave w computes hidden cols [16w, 16w+16)
        const int n0 = wave * 16;
        v8f c = {};
        const float* zr = ztraj + (size_t)(m0 + l16) * L_;
        for (int k0 = 0; k0 < L_; k0 += 4) {
            const int ka = k0 + 2 * half;
            v2f a = *(const v2f*)(zr + ka);
            v2f bb;
            bb.x = w1[(size_t)ka       * E_ + n0 + l16];
            bb.y = w1[(size_t)(ka + 1) * E_ + n0 + l16];
            c = __builtin_amdgcn_wmma_f32_16x16x4_f32(false, a, false, bb,
                                                      (short)0, c, false, false);
        }
        const float bias = b1[n0 + l16];
#pragma unroll
        for (int r = 0; r < 8; ++r) {
            const int m = r + 8 * half;
            hid[m * 257 + n0 + l16] = fmaxf(c[r] + bias, 0.f);
        }
    }
    __syncthreads();
    if (tid < 16 * S_) { // stage 2: 272 threads, one per (sample, s)
        const int m = tid / S_, s = tid % S_;
        float acc = b2[s];
        for (int e = 0; e < E_; ++e)
            acc += hid[m * 257 + e] * w2[e * S_ + s];
        out[(size_t)(m0 + m) * S_ + s] = acc;
    }
}

// ---------------------------------------------------------------------------
extern "C" void kernel_launch(void* const* d_in, const int* in_sizes, int n_in,
                              void* d_out, int out_size, void* d_ws, size_t ws_size,
                              hipStream_t stream)
{
    const float* x_history = (const float*)d_in[0];
    const float* u_history = (const float*)d_in[1];
    const float* x_init    = (const float*)d_in[2];
    /* d_in[3] x_future unused; d_in[5] n_steps == 128 (compile-time) */
    const float* u_future  = (const float*)d_in[4];
    const float* gru_wih   = (const float*)d_in[6];
    const float* gru_whh   = (const float*)d_in[7];
    const float* gru_bih   = (const float*)d_in[8];
    const float* gru_bhh   = (const float*)d_in[9];
    const float* enc_w1    = (const float*)d_in[10];
    const float* enc_b1    = (const float*)d_in[11];
    const float* enc_w2    = (const float*)d_in[12];
    const float* enc_b2    = (const float*)d_in[13];
    const float* dec_w1    = (const float*)d_in[14];
    const float* dec_b1    = (const float*)d_in[15];
    const float* dec_w2    = (const float*)d_in[16];
    const float* dec_b2    = (const float*)d_in[17];
    const float* A_w       = (const float*)d_in[18];
    const float* A_b       = (const float*)d_in[19];
    const float* B_w       = (const float*)d_in[20];
    const float* B_b       = (const float*)d_in[21];
    float* out = (float*)d_out;

    float* ws = (float*)d_ws;
    size_t off = 0;
    float* h      = ws + off; off += (size_t)B_ * H_;
    float* xpad   = ws + off; off += (size_t)B_ * T_ * KPAD;
    float* wihp   = ws + off; off += (size_t)G_ * KPAD;
    float* params = ws + off; off += (size_t)B_ * PCOLS;
    float* M1     = ws + off; off += (size_t)B_ * L_ * L_;
    float* M2     = ws + off; off += (size_t)B_ * L_ * L_;  // becomes A_dt
    float* Bmat   = ws + off; off += (size_t)B_ * L_ * C_;
    float* henc   = ws + off; off += (size_t)B_ * E_;
    float* ztraj  = ws + off; off += (size_t)B_ * TSTEPS * L_;

    const int tpb = 256;
    pack_ctxt<<<(B_ * T_ * KPAD + tpb - 1) / tpb, tpb, 0, stream>>>(x_history, u_history, xpad);
    pack_wih <<<(G_ * KPAD + tpb - 1) / tpb, tpb, 0, stream>>>(gru_wih, wihp);
    zero_f   <<<(B_ * H_ + tpb - 1) / tpb, tpb, 0, stream>>>(h, B_ * H_);

    // ---- GRU: 64 sequential fused steps (1 launch each, gi/gh in LDS) ----
    for (int t = 0; t < T_; ++t)
        gru_step<<<B_ / 16, 256, 0, stream>>>(xpad + t * KPAD, wihp, gru_whh,
                                              gru_bih, gru_bhh, h);

    // ---- params = h @ A_w + A_b  (1024 x 16512, K=256) ----
    gemm16<1><<<dim3(PCOLS / 64, B_ / 16), 128, 0, stream>>>(
        h, H_, A_w, PCOLS, A_b, params, PCOLS, B_, PCOLS, H_);
    assemble<<<(int)(((size_t)B_ * L_ * L_) / tpb), tpb, 0, stream>>>(params, M1, M2);
    solve128<<<B_, 128, 0, stream>>>(M1, M2);

    // ---- B_mat = h @ B_w + B_b ----
    gemm16<1><<<dim3((L_ * C_) / 64, B_ / 16), 128, 0, stream>>>(
        h, H_, B_w, L_ * C_, B_b, Bmat, L_ * C_, B_, L_ * C_, H_);

    // ---- encoder: z0 -> ztraj[:,0,:] ----
    enc1<<<(B_ * E_ + tpb - 1) / tpb, tpb, 0, stream>>>(x_init, enc_w1, enc_b1, henc);
    gemm16<1><<<dim3(L_ / 64, B_ / 16), 128, 0, stream>>>(
        henc, E_, enc_w2, L_, enc_b2, ztraj, TSTEPS * L_, B_, L_, E_);

    // ---- 128-step rollout, A_dt resident in LDS ----
    rollout<<<B_, 128, 0, stream>>>(M2, Bmat, u_future, ztraj);

    // ---- fused decoder MLP -> d_out ----
    decoder<<<(B_ * TSTEPS) / 16, 512, 0, stream>>>(ztraj, dec_w1, dec_b1,
                                                    dec_w2, dec_b2, out);
    (void)in_sizes; (void)n_in; (void)out_size; (void)ws_size;
}